// CustomLoss_9620726743771
// MI455X (gfx1250) — compile-verified
//
#include <hip/hip_runtime.h>
#include <math.h>

// ---------------------------------------------------------------------------
// CustomLoss (EMD via auction + KNN repulsion) for MI455X / gfx1250 (wave32).
//
// Strategy:
//  * Kernel 1: cost[b,i,j] = ||pred_i - target_j||^2 via V_WMMA_F32_16X16X4_F32
//    (K=3 padded to 4). f32 WMMA chosen because the auction assignment is
//    ordering-sensitive; fp16/fp8 would perturb the matching.
//    Cost matrix = 8*2048*2048*4B = 134MB -> resident in MI455X's 192MB L2,
//    so the 48 auction sweeps (~6.4GB of re-reads) run at L2 bandwidth.
//  * Kernel 2: per-batch auction (1 workgroup of 1024 = 32 waves per batch).
//    Per-wave top-2 row scan with shfl_xor merge (coalesced column reads),
//    per-column winner via packed u64 ds_max (deterministic: key contains row).
//    Sequential greedy fix pass with packed u64 ds_min argmin.
//    Ends with deterministic tree-reduced EMD partial per batch.
//  * Kernel 3: KNN(5) smallest self-distances per point (top-k order matched),
//    repulsion partial per block (tree reduction).
//  * Kernel 4: combine -> out[0]=emd*100, out[1]=ALPHA*rep.
// ---------------------------------------------------------------------------

#define B_SZ 8
#define NPTS 2048
#define MAX_ITERS_C 48
#define AUC_THREADS 1024
#define REP_THREADS 256
#define REP_BLOCKS ((B_SZ * NPTS) / REP_THREADS)   // 64

typedef __attribute__((ext_vector_type(2))) float v2f;
typedef __attribute__((ext_vector_type(8))) float v8f;

__device__ __forceinline__ unsigned ordf(float f) {
  unsigned u = __float_as_uint(f);
  return (u & 0x80000000u) ? ~u : (u | 0x80000000u);   // monotone float->uint
}
__device__ __forceinline__ float unordf(unsigned e) {
  unsigned u = (e & 0x80000000u) ? (e & 0x7fffffffu) : ~e;
  return __uint_as_float(u);
}

// ---------------------------------------------------------------------------
// Kernel 1: cost matrix via WMMA f32 16x16x4.  One wave -> one 16x16 tile.
// A (16x4 f32): lane half (lane>>4) selects K={0,1} vs K={2,3}; row = lane&15.
// B (4x16 f32): symmetric layout, col = lane&15.
// C/D: VGPR v, row = v + 8*(lane>>4), col = lane&15.
// ---------------------------------------------------------------------------
__global__ void __launch_bounds__(128)
cost_wmma_kernel(const float* __restrict__ pred,
                 const float* __restrict__ target,
                 float* __restrict__ cost) {
  const int lane = threadIdx.x & 31;
  const int wave = threadIdx.x >> 5;
  const int jt = blockIdx.x;                 // column tile
  const int it = blockIdx.y * 4 + wave;      // row tile
  const int b  = blockIdx.z;
  const int half = lane >> 4;
  const int m = lane & 15;

  const int row0 = it * 16;
  const int col0 = jt * 16;
  const float* P = pred   + ((size_t)b * NPTS + row0) * 3;
  const float* T = target + ((size_t)b * NPTS + col0) * 3;

  const float p0 = P[m * 3 + 0], p1 = P[m * 3 + 1], p2 = P[m * 3 + 2];
  const float t0 = T[m * 3 + 0], t1 = T[m * 3 + 1], t2 = T[m * 3 + 2];

  v2f A, Bv;
  A.x  = half ? p2 : p0;   A.y  = half ? 0.0f : p1;   // K = 2*half, 2*half+1
  Bv.x = half ? t2 : t0;   Bv.y = half ? 0.0f : t1;

  v8f C = {};
  // D = A(16x4) x B(4x16): D[v] = dot(pred[row0+v+8*half], target[col0+m])
  C = __builtin_amdgcn_wmma_f32_16x16x4_f32(false, A, false, Bv,
                                            (short)0, C, false, false);

  const float bb = t0 * t0 + t1 * t1 + t2 * t2;   // ||target_col||^2
  float* cb = cost + (size_t)b * NPTS * NPTS;
#pragma unroll
  for (int v = 0; v < 8; ++v) {
    const int row = row0 + v + 8 * half;
    const float* pr = pred + ((size_t)b * NPTS + row) * 3;
    const float aa = pr[0] * pr[0] + pr[1] * pr[1] + pr[2] * pr[2];
    cb[(size_t)row * NPTS + col0 + m] = aa + bb - 2.0f * C[v];
  }
}

// ---------------------------------------------------------------------------
// Kernel 2: auction assignment + EMD partial.  One block (1024 thr) per batch.
// ---------------------------------------------------------------------------
__global__ void __launch_bounds__(AUC_THREADS)
auction_kernel(const float* __restrict__ pred,
               const float* __restrict__ target,
               const float* __restrict__ cost_all,
               float* __restrict__ emd_sum) {
  const int b   = blockIdx.x;
  const int tid = threadIdx.x;
  const int wv  = tid >> 5;
  const int ln  = tid & 31;
  const float* cost = cost_all + (size_t)b * NPTS * NPTS;

  __shared__ float prices[NPTS];
  __shared__ int   owner[NPTS];
  __shared__ int   assg[NPTS];
  __shared__ unsigned long long bidpack[NPTS];
  __shared__ int   owned_s[NPTS];
  __shared__ float sred[AUC_THREADS];
  __shared__ unsigned long long minp;
  __shared__ int   flag;
  __shared__ float eps_sh;

  for (int i = tid; i < NPTS; i += AUC_THREADS) {
    prices[i] = 0.0f; owner[i] = -1; assg[i] = -1;
  }

  // eps_bid = max(max(cost), 1e-6)/64  (fixed-order tree reduction)
  float mx = -__builtin_inff();
  for (size_t k = tid; k < (size_t)NPTS * NPTS; k += AUC_THREADS)
    mx = fmaxf(mx, cost[k]);
  sred[tid] = mx;
  __syncthreads();
  for (int s = AUC_THREADS / 2; s > 0; s >>= 1) {
    if (tid < s) sred[tid] = fmaxf(sred[tid], sred[tid + s]);
    __syncthreads();
  }
  if (tid == 0) eps_sh = fmaxf(sred[0], 1e-6f) / 64.0f;
  __syncthreads();
  const float eps_bid = eps_sh;

  for (int it = 0; it < MAX_ITERS_C; ++it) {
    if (tid == 0) flag = 0;
    __syncthreads();
    for (int i = tid; i < NPTS; i += AUC_THREADS)
      if (assg[i] < 0) flag = 1;
    for (int j = tid; j < NPTS; j += AUC_THREADS) bidpack[j] = 0ull;
    __syncthreads();
    if (flag == 0) break;

    // --- bidding: one wave per row, lanes scan coalesced columns ---
    for (int i = wv; i < NPTS; i += 32) {
      if (assg[i] >= 0) continue;                       // wave-uniform
      const float* crow = cost + (size_t)i * NPTS;
      float v1 = -__builtin_inff(), v2 = -__builtin_inff();
      int j1 = 0x7fffffff;
      for (int j = ln; j < NPTS; j += 32) {
        const float v = -crow[j] - prices[j];
        if (v > v1)      { v2 = v1; v1 = v; j1 = j; }
        else if (v > v2) { v2 = v; }
      }
      // cross-lane top-2 merge (tie-break: lowest index, matches top_k)
      for (int off = 16; off > 0; off >>= 1) {
        const float ov1 = __shfl_xor(v1, off);
        const float ov2 = __shfl_xor(v2, off);
        const int   oj1 = __shfl_xor(j1, off);
        const bool take = (ov1 > v1) || (ov1 == v1 && oj1 < j1);
        if (take) { v2 = fmaxf(ov2, v1); v1 = ov1; j1 = oj1; }
        else      { v2 = fmaxf(v2, ov1); }
      }
      if (ln == 0) {
        const float bid = prices[j1] + (v1 - v2) + eps_bid;
        const unsigned long long pk =
            ((unsigned long long)ordf(bid) << 32) |
            (unsigned long long)(~(unsigned)i);        // max pk -> lowest row on tie
        atomicMax(&bidpack[j1], pk);
      }
    }
    __syncthreads();

    // --- assignment: prev owners (assigned) and winners (unassigned) disjoint ---
    for (int j = tid; j < NPTS; j += AUC_THREADS) {
      const unsigned long long pk = bidpack[j];
      if (pk != 0ull) {
        const int winner = (int)(~(unsigned)(pk & 0xffffffffull));
        const float wb = unordf((unsigned)(pk >> 32));
        const int prev = owner[j];
        if (prev >= 0) assg[prev] = -1;
        assg[winner] = j;
        owner[j] = winner;
        prices[j] = wb;
      }
    }
    __syncthreads();
  }

  // --- greedy fix pass (sequential over rows, matches reference scan) ---
  for (int j = tid; j < NPTS; j += AUC_THREADS) owned_s[j] = (owner[j] >= 0);
  __syncthreads();
  for (int i = 0; i < NPTS; ++i) {
    const int need = (assg[i] < 0);                    // uniform
    if (need) {
      if (tid == 0) minp = ~0ull;
      __syncthreads();
      const float* crow = cost + (size_t)i * NPTS;
      for (int j = tid; j < NPTS; j += AUC_THREADS) {
        if (!owned_s[j]) {
          const unsigned long long pk =
              ((unsigned long long)ordf(crow[j]) << 32) | (unsigned)j;
          atomicMin(&minp, pk);
        }
      }
      __syncthreads();
      if (tid == 0) {
        const unsigned long long pk = minp;
        const int j = (pk == ~0ull) ? 0 : (int)(unsigned)(pk & 0xffffffffull);
        assg[i] = j;
        owned_s[j] = 1;
      }
      __syncthreads();
    }
  }

  // --- EMD partial: sum_i ||pred_i - target_assg[i]||^2 ---
  float acc = 0.0f;
  for (int i = tid; i < NPTS; i += AUC_THREADS) {
    const int j = assg[i];
    const float* p = pred   + ((size_t)b * NPTS + i) * 3;
    const float* t = target + ((size_t)b * NPTS + j) * 3;
    const float dx = p[0] - t[0], dy = p[1] - t[1], dz = p[2] - t[2];
    acc += dx * dx + dy * dy + dz * dz;
  }
  sred[tid] = acc;
  __syncthreads();
  for (int s = AUC_THREADS / 2; s > 0; s >>= 1) {
    if (tid < s) sred[tid] += sred[tid + s];
    __syncthreads();
  }
  if (tid == 0) emd_sum[b] = sred[0];
}

// ---------------------------------------------------------------------------
// Kernel 3: repulsion.  One thread per (b,i); keep 5 smallest d^2 (stable
// insertion matches top_k tie-breaking); drop slot 0 (self), use slots 1..4.
// ---------------------------------------------------------------------------
__global__ void __launch_bounds__(REP_THREADS)
rep_kernel(const float* __restrict__ pred, float* __restrict__ rep_part) {
  const int gid = blockIdx.x * REP_THREADS + threadIdx.x;
  const int b = gid / NPTS;
  const int i = gid % NPTS;
  const float* P = pred + (size_t)b * NPTS * 3;
  const float px = P[i * 3 + 0], py = P[i * 3 + 1], pz = P[i * 3 + 2];

  float bd[5];
#pragma unroll
  for (int k = 0; k < 5; ++k) bd[k] = __builtin_inff();

  for (int j = 0; j < NPTS; ++j) {
    const float dx = P[j * 3 + 0] - px;
    const float dy = P[j * 3 + 1] - py;
    const float dz = P[j * 3 + 2] - pz;
    const float d2 = dx * dx + dy * dy + dz * dz;
    if (d2 < bd[4]) {
      int k = 4;
      while (k > 0 && d2 < bd[k - 1]) { bd[k] = bd[k - 1]; --k; }
      bd[k] = d2;
    }
  }

  float acc = 0.0f;
#pragma unroll
  for (int k = 1; k < 5; ++k) {                 // skip self (slot 0)
    const float d2 = fmaxf(bd[k], 1e-12f);
    const float dr = sqrtf(d2);
    acc += (0.07f - dr) * expf(-d2 / (0.03f * 0.03f));
  }

  __shared__ float sred[REP_THREADS];
  sred[threadIdx.x] = acc;
  __syncthreads();
  for (int s = REP_THREADS / 2; s > 0; s >>= 1) {
    if (threadIdx.x < s) sred[threadIdx.x] += sred[threadIdx.x + s];
    __syncthreads();
  }
  if (threadIdx.x == 0) rep_part[blockIdx.x] = sred[0];
}

// ---------------------------------------------------------------------------
// Kernel 4: finalize the two scalars.
// ---------------------------------------------------------------------------
__global__ void final_kernel(const float* __restrict__ emd_sum,
                             const float* __restrict__ rep_part,
                             const float* __restrict__ pcd_radius,
                             float* __restrict__ out) {
  if (threadIdx.x == 0 && blockIdx.x == 0) {
    float e = 0.0f;
    for (int b = 0; b < B_SZ; ++b)
      e += (emd_sum[b] / ((float)NPTS * 3.0f)) / pcd_radius[b];
    e /= (float)B_SZ;
    float r = 0.0f;
    for (int k = 0; k < REP_BLOCKS; ++k) r += rep_part[k];
    r /= ((float)B_SZ * (float)NPTS * 4.0f);
    out[0] = e * 100.0f;        // emd * 100
    out[1] = 1.0f * r;          // ALPHA * rep
  }
}

// ---------------------------------------------------------------------------
extern "C" void kernel_launch(void* const* d_in, const int* in_sizes, int n_in,
                              void* d_out, int out_size, void* d_ws, size_t ws_size,
                              hipStream_t stream) {
  (void)in_sizes; (void)n_in; (void)out_size; (void)ws_size;

  const float* pred   = (const float*)d_in[0];
  const float* target = (const float*)d_in[1];
  const float* pcd    = (const float*)d_in[2];
  float* out = (float*)d_out;

  // workspace layout: [cost 8*2048*2048 f32][emd_sum 8 f32][rep_part 64 f32]
  float* cost     = (float*)d_ws;
  float* emd_sum  = cost + (size_t)B_SZ * NPTS * NPTS;
  float* rep_part = emd_sum + B_SZ;

  dim3 g1(NPTS / 16, NPTS / 16 / 4, B_SZ);     // 128 x 32 x 8, 4 waves/block
  cost_wmma_kernel<<<g1, 128, 0, stream>>>(pred, target, cost);
  auction_kernel<<<B_SZ, AUC_THREADS, 0, stream>>>(pred, target, cost, emd_sum);
  rep_kernel<<<REP_BLOCKS, REP_THREADS, 0, stream>>>(pred, rep_part);
  final_kernel<<<1, 32, 0, stream>>>(emd_sum, rep_part, pcd, out);
}